// RGCN_69861938037148
// MI455X (gfx1250) — compile-verified
//
#include <hip/hip_runtime.h>
#include <hip/hip_bf16.h>

// ---------------------------------------------------------------------------
// RGCN layer for MI455X (gfx1250, wave32, WMMA).
//
// Strategy (mean-aggregate is linear-commutative):
//   acc[t][r][:]  = sum over edges e with dst=t, type=r of x_src[src(e)]
//   cnt[t][r]     = number of such edges
//   out[t]        = sum_r (acc[t][r]/max(cnt,1)) @ rel_w[r]^T
//                 + x_target[t] @ root_w[type(t)]^T + root_b[type(t)]
//
// Workspace layout (fp32):
//   [0 .. NT*7*128)            acc
//   [NT*7*128 .. +NT*7)        cnt
// Total = 50000*7*128 + 50000*7 floats = 180.6 MB.
// ---------------------------------------------------------------------------

#define N_TGT_C   50000
#define IN_CH_C   128
#define OUT_CH_C  128
#define NREL_C    7

typedef __attribute__((ext_vector_type(2))) float v2f;
typedef __attribute__((ext_vector_type(8))) float v8f;

// ---------------- Kernel 1: zero the workspace accumulators ----------------
__global__ void rgcn_zero_ws(float* __restrict__ ws, long n_floats) {
    long i = (long)blockIdx.x * blockDim.x + threadIdx.x;
    long stride = (long)gridDim.x * blockDim.x;
    // vectorized zero: n_floats is a multiple of 4 by construction
    long n4 = n_floats >> 2;
    float4* p = (float4*)ws;
    for (long j = i; j < n4; j += stride) {
        p[j] = make_float4(0.f, 0.f, 0.f, 0.f);
    }
}

// ---------------- Kernel 2: per-edge scatter-add (one wave32 per edge) -----
__global__ void rgcn_scatter(const float* __restrict__ x_src,
                             const int* __restrict__ edge_src,
                             const int* __restrict__ edge_dst,
                             const int* __restrict__ edge_type,
                             float* __restrict__ acc,
                             float* __restrict__ cnt,
                             int n_edges) {
    int gid  = blockIdx.x * blockDim.x + threadIdx.x;
    int e    = gid >> 5;           // one wave per edge
    int lane = threadIdx.x & 31;
    if (e >= n_edges) return;

    int s = edge_src[e];
    int d = edge_dst[e];
    int r = edge_type[e];

    // 32 lanes x float4 = 128 channels
    const float4 v = ((const float4*)(x_src + (long)s * IN_CH_C))[lane];
    float* a = acc + ((long)d * NREL_C + r) * IN_CH_C + lane * 4;
    atomicAdd(a + 0, v.x);
    atomicAdd(a + 1, v.y);
    atomicAdd(a + 2, v.z);
    atomicAdd(a + 3, v.w);
    if (lane == 0) atomicAdd(cnt + (long)d * NREL_C + r, 1.0f);
}

// ---------------- Kernel 3: relation GEMM via V_WMMA_F32_16X16X4_F32 -------
// Block = 256 threads = 8 waves. Block b covers target rows [16b, 16b+16).
// Wave w covers output columns [16w, 16w+16). Full fp32 precision.
//
// ISA 32-bit A 16x4 layout : lane(0-15)=row M, VGPR0/1 = K0/K1 (lanes 16-31: K2/K3)
// ISA 32-bit B 4x16 layout : lane(0-15)=col N, VGPR0/1 = K0/K1 (lanes 16-31: K2/K3)
// ISA 32-bit C/D 16x16     : lane(0-15)=col N, VGPR v = row v (lanes 16-31: row v+8)
__global__ void rgcn_gemm(const float* __restrict__ acc,
                          const float* __restrict__ cnt,
                          const float* __restrict__ rel_w,
                          float* __restrict__ out) {
    const int tile = blockIdx.x;            // 0 .. 3124 (50000/16)
    const int wave = threadIdx.x >> 5;      // 0 .. 7  (N-tile)
    const int lane = threadIdx.x & 31;
    const int l    = lane & 15;
    const int half = lane >> 4;             // selects K-pair {0,1} vs {2,3}

    const int rowA = tile * 16 + l;         // this lane's A row (M)
    const int colB = wave * 16 + l;         // this lane's B column (N = output ch)

    v8f c = {};
    for (int r = 0; r < NREL_C; ++r) {
        const float cn    = cnt[(long)rowA * NREL_C + r];
        const float scale = 1.0f / fmaxf(cn, 1.0f);
        const float* Ab = acc + ((long)rowA * NREL_C + r) * IN_CH_C + half * 2;
        const float* Bb = rel_w + ((long)r * OUT_CH_C + colB) * IN_CH_C + half * 2;
        #pragma unroll
        for (int kk = 0; kk < IN_CH_C; kk += 4) {
            v2f a, b;
            a.x = Ab[kk]     * scale;
            a.y = Ab[kk + 1] * scale;
            b.x = Bb[kk];
            b.y = Bb[kk + 1];
            c = __builtin_amdgcn_wmma_f32_16x16x4_f32(
                    /*neg_a=*/false, a, /*neg_b=*/false, b,
                    /*c_mod=*/(short)0, c, /*reuse_a=*/false, /*reuse_b=*/false);
        }
    }

    #pragma unroll
    for (int v = 0; v < 8; ++v) {
        int row = tile * 16 + v + half * 8;
        out[(long)row * OUT_CH_C + wave * 16 + l] = c[v];
    }
}

// ---------------- Kernel 4: root linear (per-row weight select) ------------
// One thread per output element; 128-MAC fp32 dot with float4 loads.
__global__ void rgcn_root(const float* __restrict__ x_target,
                          const float* __restrict__ root_w,
                          const float* __restrict__ root_b,
                          const int* __restrict__ ttype,
                          float* __restrict__ out) {
    int idx = blockIdx.x * blockDim.x + threadIdx.x;   // t*128 + o
    int t = idx >> 7;
    int o = idx & 127;
    if (t >= N_TGT_C) return;
    int tt = ttype[t];
    const float4* xr = (const float4*)(x_target + (long)t * IN_CH_C);
    const float4* wr = (const float4*)(root_w + ((long)tt * OUT_CH_C + o) * IN_CH_C);
    float s = 0.0f;
    #pragma unroll 8
    for (int k = 0; k < IN_CH_C / 4; ++k) {
        float4 a = xr[k];
        float4 b = wr[k];
        s += a.x * b.x + a.y * b.y + a.z * b.z + a.w * b.w;
    }
    out[idx] += s + root_b[(long)tt * OUT_CH_C + o];
}

// ---------------------------------------------------------------------------
extern "C" void kernel_launch(void* const* d_in, const int* in_sizes, int n_in,
                              void* d_out, int out_size, void* d_ws, size_t ws_size,
                              hipStream_t stream) {
    const float* x_src    = (const float*)d_in[0];
    const float* x_target = (const float*)d_in[1];
    const float* rel_w    = (const float*)d_in[2];
    const float* root_w   = (const float*)d_in[3];
    const float* root_b   = (const float*)d_in[4];
    const int*   edge_src = (const int*)d_in[5];
    const int*   edge_dst = (const int*)d_in[6];
    const int*   edge_typ = (const int*)d_in[7];
    const int*   ttype    = (const int*)d_in[8];
    float*       out      = (float*)d_out;

    const int n_edges = in_sizes[5];
    const int n_tgt   = in_sizes[8];          // 50000

    float* acc = (float*)d_ws;                                    // NT*7*128
    float* cnt = acc + (long)n_tgt * NREL_C * IN_CH_C;            // NT*7
    const long ws_floats = (long)n_tgt * NREL_C * IN_CH_C + (long)n_tgt * NREL_C;
    // pad zeroing region up to multiple of 4 floats
    const long ws_floats4 = (ws_floats + 3) & ~3L;

    // 1) zero accumulators + counts
    rgcn_zero_ws<<<2048, 256, 0, stream>>>((float*)d_ws, ws_floats4);

    // 2) edge scatter (one wave32 per edge, 8 edges per 256-thread block)
    int sblocks = (n_edges + 7) / 8;
    rgcn_scatter<<<sblocks, 256, 0, stream>>>(x_src, edge_src, edge_dst, edge_typ,
                                              acc, cnt, n_edges);

    // 3) relation GEMM with fp32 WMMA (writes every element of out)
    rgcn_gemm<<<n_tgt / 16, 256, 0, stream>>>(acc, cnt, rel_w, out);

    // 4) root linear, accumulates into out
    rgcn_root<<<(n_tgt * OUT_CH_C) / 256, 256, 0, stream>>>(x_target, root_w, root_b,
                                                            ttype, out);
}